// DigitCaps_48112223650521
// MI455X (gfx1250) — compile-verified
//
#include <hip/hip_runtime.h>

typedef __attribute__((ext_vector_type(2))) float v2f;
typedef __attribute__((ext_vector_type(8))) float v8f;

#define NTILES 10          // 160 output cols / 16
#define RTOT   4608
#define ITOT   8
#define KTOT   (RTOT * ITOT)
#define CODIM  160
#define LDSTR  164         // padded LDS row stride (floats) to avoid bank conflicts

// ---------------------------------------------------------------------------
// Kernel 1: split-K GEMM partials.
// grid.x = 8 (M-tiles of 16 batch rows) * KCHUNKS.  256 threads = 8 wave32.
// Each wave accumulates all 10 N-tiles (160 cols) over its K-slice with
// V_WMMA_F32_16X16X4_F32, then waves combine via ds_add_f32 in LDS, and the
// workgroup writes its [16 x 160] partial block to workspace.
// ---------------------------------------------------------------------------
__global__ __launch_bounds__(256) void caps_gemm_partial(
    const float* __restrict__ x, const float* __restrict__ W,
    float* __restrict__ ws, int rPerChunk)
{
    __shared__ float lds[16 * LDSTR];

    const int wg     = blockIdx.x;
    const int m_t    = wg & 7;        // which 16-row batch tile
    const int chunk  = wg >> 3;       // which K chunk
    const int tid    = threadIdx.x;
    const int wave   = tid >> 5;
    const int lane   = tid & 31;
    const int hi     = lane >> 4;     // upper half-wave handles K+2 lanes
    const int lane_o = lane & 15;

    // zero the LDS accumulation buffer (including padding)
    for (int j = tid; j < 16 * LDSTR; j += 256) lds[j] = 0.0f;
    __syncthreads();

    const int rPerWave = rPerChunk >> 3;           // 8 waves split the chunk
    const int rBeg = chunk * rPerChunk + wave * rPerWave;
    const int rEnd = rBeg + rPerWave;

    // A fragment: lane_o selects the batch row (M), hi selects K-pair {0,1} vs {2,3}
    const int row = m_t * 16 + lane_o;
    const float* xrow = x + (size_t)row * KTOT + 2 * hi;

    v8f acc[NTILES];
#pragma unroll
    for (int nt = 0; nt < NTILES; ++nt) acc[nt] = (v8f)0.0f;

    for (int r = rBeg; r < rEnd; ++r) {
        const float* xp = xrow + r * ITOT;
        v2f a0 = *(const v2f*)(xp);        // K = r*8 + {0..3}
        v2f a1 = *(const v2f*)(xp + 4);    // K = r*8 + {4..7}

        // B fragment: lane_o selects output col (o), hi selects K-pair
        const float* wp = W + (size_t)r * (CODIM * ITOT) + lane_o * ITOT + 2 * hi;
#pragma unroll
        for (int nt = 0; nt < NTILES; ++nt) {
            v2f b0 = *(const v2f*)(wp + nt * (16 * ITOT));
            v2f b1 = *(const v2f*)(wp + nt * (16 * ITOT) + 4);
            acc[nt] = __builtin_amdgcn_wmma_f32_16x16x4_f32(
                false, a0, false, b0, (short)0, acc[nt], false, false);
            acc[nt] = __builtin_amdgcn_wmma_f32_16x16x4_f32(
                false, a1, false, b1, (short)0, acc[nt], false, false);
        }
    }

    // Cross-wave reduction: C/D layout puts (M = v + 8*hi, N = lane_o) in acc[nt][v]
#pragma unroll
    for (int nt = 0; nt < NTILES; ++nt) {
#pragma unroll
        for (int v = 0; v < 8; ++v) {
            const int mloc = v + 8 * hi;
            atomicAdd(&lds[mloc * LDSTR + nt * 16 + lane_o], acc[nt][v]);
        }
    }
    __syncthreads();

    // Coalesced write of this WG's partial [16 x 160] block
    float* out = ws + (size_t)(chunk * 8 + m_t) * (16 * CODIM);
    for (int j = tid; j < 16 * CODIM; j += 256) {
        const int m = j / CODIM;
        const int col = j - m * CODIM;
        out[j] = lds[m * LDSTR + col];
    }
}

// ---------------------------------------------------------------------------
// Kernel 2: reduce K-chunk partials, scale by 1/R, squash over o (16 lanes).
// Flat element f maps to (b, c, o) with o innermost; the 16 threads sharing
// (b,c) live in one half-wave, so xor-shuffles with masks < 16 stay in-group.
// ---------------------------------------------------------------------------
__global__ __launch_bounds__(256) void caps_reduce_squash(
    const float* __restrict__ ws, float* __restrict__ out, int chunks)
{
    const int f = blockIdx.x * 256 + threadIdx.x;   // 0 .. 20479
    float s = 0.0f;
    for (int k = 0; k < chunks; ++k)
        s += ws[(size_t)k * 20480 + f];
    s *= (1.0f / (float)RTOT);

    float sq = s * s;
    sq += __shfl_xor(sq, 1, 32);
    sq += __shfl_xor(sq, 2, 32);
    sq += __shfl_xor(sq, 4, 32);
    sq += __shfl_xor(sq, 8, 32);

    // sq_norm * s / ((1+sq_norm)*sqrt(sq_norm)) == s * sqrt(sq_norm) / (1+sq_norm)
    out[f] = s * sqrtf(sq) / (1.0f + sq);
}

extern "C" void kernel_launch(void* const* d_in, const int* in_sizes, int n_in,
                              void* d_out, int out_size, void* d_ws, size_t ws_size,
                              hipStream_t stream) {
    (void)in_sizes; (void)n_in; (void)out_size;
    const float* x = (const float*)d_in[0];   // [128, 4608, 8]
    const float* W = (const float*)d_in[1];   // [1, 4608, 10, 16, 8]
    float* out = (float*)d_out;               // [128, 10, 16]
    float* ws  = (float*)d_ws;

    // Pick the K-split: chunks must divide 4608 with rPerChunk % 8 == 0,
    // and partials (chunks * 20480 floats) must fit in the workspace.
    static const int cand[] = {72, 36, 18, 12, 9, 6, 4, 3, 2, 1};
    int chunks = 1;
    for (int i = 0; i < 10; ++i) {
        if ((size_t)cand[i] * 20480u * sizeof(float) <= ws_size) { chunks = cand[i]; break; }
    }
    const int rPerChunk = RTOT / chunks;

    caps_gemm_partial<<<dim3(8 * chunks), dim3(256), 0, stream>>>(x, W, ws, rPerChunk);
    caps_reduce_squash<<<dim3(20480 / 256), dim3(256), 0, stream>>>(ws, out, chunks);
}